// HSTUBlockPostprocessor_67680094650967
// MI455X (gfx1250) — compile-verified
//
#include <hip/hip_runtime.h>
#include <stdint.h>

// ---------------------------------------------------------------------------
// HSTU block postprocessor: gather candidate-tail rows (every 2nd candidate)
// and L2-normalize each 256-float row.  Pure bandwidth problem (~268 MB,
// ~11.5us at 23.3 TB/s) -> CDNA5 async global->LDS DMA (ASYNCcnt path),
// double-buffered per wave32, scalar (SALU/SMEM) segment lookup, wave32
// shuffle reduction, b128 nontemporal stores.  No WMMA: arithmetic intensity
// ~0.37 flop/byte, far below the matrix-op roofline knee.
// ---------------------------------------------------------------------------

typedef __attribute__((ext_vector_type(4))) float f32x4;
typedef __attribute__((ext_vector_type(4))) int   i32x4;

#define AS1 __attribute__((address_space(1)))
#define AS3 __attribute__((address_space(3)))

#if defined(__HIP_DEVICE_COMPILE__) && __has_builtin(__builtin_amdgcn_global_load_async_to_lds_b128)
#define HAVE_ASYNC 1
#else
#define HAVE_ASYNC 0
#endif

#if defined(__HIP_DEVICE_COMPILE__) && __has_builtin(__builtin_amdgcn_s_wait_asynccnt)
#define WAIT_ASYNC(n) __builtin_amdgcn_s_wait_asynccnt(n)
#else
#define WAIT_ASYNC(n) asm volatile("s_wait_asynccnt " #n ::: "memory")
#endif

constexpr int D = 256;               // floats per row (1 KB)
constexpr int ROWS_PER_WAVE = 8;     // rows handled by one wave32
constexpr int WAVES_PER_BLOCK = 8;   // 256 threads/block

__global__ __launch_bounds__(256) void hstu_gather_l2norm_kernel(
    const float* __restrict__ values,
    const int*   __restrict__ seq_off,   // B+1 entries
    const int*   __restrict__ cand_off,  // B+1 entries
    float*       __restrict__ out,
    int nrows,                           // total output rows = total_candidates/2
    int nseg)                            // B
{
#if HAVE_ASYNC
    // 2 x 1KB double buffer per wave
    __shared__ float lds[WAVES_PER_BLOCK * 2 * D];
#endif
    const int  lane = threadIdx.x & 31;
    // readfirstlane: make the wave id (and everything derived from it)
    // provably uniform so the offset-table walk becomes SALU + s_load
    // instead of per-lane VMEM.
    const int  wave = __builtin_amdgcn_readfirstlane(threadIdx.x >> 5);
    const long wgid = (long)blockIdx.x * WAVES_PER_BLOCK + wave;
    const long row0 = wgid * ROWS_PER_WAVE;
    if (row0 >= (long)nrows) return;
    const long rem = (long)nrows - row0;
    const int  cnt = rem < ROWS_PER_WAVE ? (int)rem : ROWS_PER_WAVE;

    // --- segment lookup: largest b with cand_off[b] <= j0  (j0 = 2*row0) ---
    const long j0 = row0 * 2;
    int lo = 0, hi = nseg - 1;
    while (lo < hi) {
        int mid = (lo + hi + 1) >> 1;
        if ((long)cand_off[mid] <= j0) lo = mid; else hi = mid - 1;
    }
    int b = lo;

    // src row for candidate index j (monotone advance of b); wave-uniform.
    auto src_row = [&](long j) -> long {
        while (j >= (long)cand_off[b + 1]) ++b;
        const int sb  = seq_off[b],  sb1 = seq_off[b + 1];
        const int cb  = cand_off[b], cb1 = cand_off[b + 1];
        // src = seq_off[b] + len_a + (j - cand_off[b]); len_a = (sb1-sb)-(cb1-cb)
        return (long)sb + (long)(sb1 - sb) - (long)(cb1 - cb) + (j - (long)cb);
    };

#if HAVE_ASYNC
    float* mybuf = &lds[wave * 2 * D];

    // Async DMA one 1KB row into LDS: 2 x b128 per lane (32 lanes x 16B = 512B
    // per instruction; imm offset applies to BOTH global and LDS addresses).
    auto issue = [&](int k) {
        const long  j  = (row0 + k) * 2;
        const long  s  = src_row(j);
        const float* gp = values + s * (long)D + lane * 4;          // 16B/lane
        float*       lp = mybuf + ((k & 1) ? D : 0) + lane * 4;
        AS1 i32x4* g3 = (AS1 i32x4*)(uintptr_t)gp;
        AS3 i32x4* l3 = (AS3 i32x4*)(uint32_t)(uintptr_t)lp;
        __builtin_amdgcn_global_load_async_to_lds_b128(g3, l3, 0,   0);
        __builtin_amdgcn_global_load_async_to_lds_b128(g3, l3, 512, 0);
    };
#endif

    // Process one row: read 8 floats/lane, wave32 butterfly reduce the sum of
    // squares, scale by 1/max(norm,1e-6), nontemporal b128 stores.
    auto process = [&](int k
#if !HAVE_ASYNC
                       , long s
#endif
                       ) {
#if HAVE_ASYNC
        const float* rp = mybuf + ((k & 1) ? D : 0) + lane * 8;     // 32B/lane
        f32x4 x0 = *(const f32x4*)(rp);
        f32x4 x1 = *(const f32x4*)(rp + 4);
#else
        const f32x4* gp = (const f32x4*)(values + s * (long)D + lane * 8);
        f32x4 x0 = __builtin_nontemporal_load(gp);
        f32x4 x1 = __builtin_nontemporal_load(gp + 1);
#endif
        float ss = 0.0f;
#pragma unroll
        for (int t = 0; t < 4; ++t) {
            ss = fmaf(x0[t], x0[t], ss);
            ss = fmaf(x1[t], x1[t], ss);
        }
#pragma unroll
        for (int m = 16; m >= 1; m >>= 1)
            ss += __shfl_xor(ss, m, 32);

        const float inv = 1.0f / fmaxf(sqrtf(ss), 1e-6f);
        f32x4 y0 = x0 * inv;
        f32x4 y1 = x1 * inv;

        float* op = out + (row0 + k) * (long)D + lane * 8;
        __builtin_nontemporal_store(y0, (f32x4*)op);
        __builtin_nontemporal_store(y1, (f32x4*)(op + 4));
    };

#if HAVE_ASYNC
    if (cnt == ROWS_PER_WAVE) {
        // Hot path: constant trip count -> fully unrolled pipeline.
        issue(0);
#pragma unroll
        for (int k = 0; k < ROWS_PER_WAVE; ++k) {
            if (k + 1 < ROWS_PER_WAVE) {
                issue(k + 1);      // prefetch next row into other buffer
                WAIT_ASYNC(2);     // in-order completion: row k's 2 ops done
            } else {
                WAIT_ASYNC(0);
            }
            process(k);
        }
    } else {
        issue(0);
        for (int k = 0; k < cnt; ++k) {
            if (k + 1 < cnt) {
                issue(k + 1);
                WAIT_ASYNC(2);
            } else {
                WAIT_ASYNC(0);
            }
            process(k);
        }
    }
#else
    for (int k = 0; k < cnt; ++k) {
        const long j = (row0 + k) * 2;
        process(k, src_row(j));
    }
#endif
}

extern "C" void kernel_launch(void* const* d_in, const int* in_sizes, int n_in,
                              void* d_out, int out_size, void* d_ws, size_t ws_size,
                              hipStream_t stream) {
    (void)d_ws; (void)ws_size; (void)n_in;
    const float* values   = (const float*)d_in[0];
    const int*   seq_off  = (const int*)d_in[1];
    const int*   cand_off = (const int*)d_in[2];
    float*       out      = (float*)d_out;

    const int nseg  = in_sizes[1] - 1;          // B
    const int nrows = out_size / D;             // total_candidates / 2
    const long waves  = ((long)nrows + ROWS_PER_WAVE - 1) / ROWS_PER_WAVE;
    const int  blocks = (int)((waves + WAVES_PER_BLOCK - 1) / WAVES_PER_BLOCK);

    hstu_gather_l2norm_kernel<<<blocks, WAVES_PER_BLOCK * 32, 0, stream>>>(
        values, seq_off, cand_off, out, nrows, nseg);
}